// CAComm_54829552501030
// MI455X (gfx1250) — compile-verified
//
#include <hip/hip_runtime.h>
#include <hip/hip_bf16.h>

// out = upsample2x_nearest(parent_x @ conv_kernel + bias) * Wv
// (attention collapses: softmax rows sum to 1, so x_new == V == s*Wv)
//
// parent_x : (4,128,128,16) f32   -> 65536 pixels x 16 ch
// output   : (4,256,256,16) f32
//
// One wave32 = one 16x16 fp32 A-tile (16 pixels x 16 in-channels),
// K=16 folded as 4 chained v_wmma_f32_16x16x4_f32 ops.

typedef __attribute__((ext_vector_type(2))) float v2f;
typedef __attribute__((ext_vector_type(8))) float v8f;

#define CH      16
#define PAR_PIX (4 * 128 * 128)   // 65536 parent pixels
#define WAVES   (PAR_PIX / 16)    // 4096 waves, 16 pixels each

__global__ __launch_bounds__(256) void cacomm_fused_kernel(
    const float* __restrict__ parent,   // (4,128,128,16)
    const float* __restrict__ ck,       // (16,16) conv kernel, row-major [cin][cout]
    const float* __restrict__ bias,     // (16,)
    const float* __restrict__ wv,       // (1,)
    float* __restrict__ out)            // (4,256,256,16)
{
    const int lane = threadIdx.x & 31;
    const int m    = lane & 15;   // A: row M / B,C,D: column N (= out channel)
    const int hi   = lane >> 4;   // lane half selects K sub-pair / row half
    const int wave = blockIdx.x * (blockDim.x >> 5) + (threadIdx.x >> 5);
    const int P0   = wave << 4;   // first parent pixel of this 16-pixel tile

    // Stream-prefetch the next tile's A data (speculative; OOB is dropped).
    __builtin_prefetch(parent + (size_t)(P0 + 16) * CH, 0, 0);

    // Accumulator seeded with bias[N] -> "+ bias" comes for free.
    const float bm = bias[m];
    v8f c;
#pragma unroll
    for (int i = 0; i < 8; ++i) c[i] = bm;

    // A-matrix 16x4 fp32 layout: lane holds (M = lane%16, K = 4*kb + 2*hi + {0,1})
    // -> one aligned float2 load per K-step (channels are contiguous per pixel).
    const float* aBase = parent + (size_t)(P0 + m) * CH + hi * 2;
#pragma unroll
    for (int kb = 0; kb < 4; ++kb) {
        v2f a = *(const v2f*)(aBase + kb * 4);
        // B-matrix 4x16 fp32: N = lane%16, K split across lane halves / 2 VGPRs.
        const int krow = kb * 4 + hi * 2;
        v2f b;
        b.x = ck[(krow + 0) * CH + m];
        b.y = ck[(krow + 1) * CH + m];
        // 8 args: (neg_a, A, neg_b, B, c_mod, C, reuse_a, reuse_b)
        c = __builtin_amdgcn_wmma_f32_16x16x4_f32(
                false, a, false, b, (short)0, c, false, false);
    }

    // Scale by Wv and scatter each value to its 2x2 nearest-upsample footprint.
    const float wv0 = wv[0];
    const int b_   = P0 >> 14;          // 128*128 = 16384 pixels per batch
    const int h    = (P0 >> 7) & 127;
    const int w0   = P0 & 127;          // tile spans w0..w0+15 within one row
    float* obase   = out + (size_t)b_ * (256 * 256 * 16);
    const int row0 = h * 2;

#pragma unroll
    for (int r = 0; r < 8; ++r) {
        const int   M    = r + hi * 8;        // pixel row within tile
        const float v    = c[r] * wv0;
        const int   col0 = (w0 + M) * 2;
#pragma unroll
        for (int dy = 0; dy < 2; ++dy) {
#pragma unroll
            for (int dx = 0; dx < 2; ++dx) {
                // lanes 0-15 and 16-31 each write a contiguous 16-float
                // channel vector -> two 64B segments per store instruction.
                obase[((size_t)(row0 + dy) * 256 + (col0 + dx)) * CH + m] = v;
            }
        }
    }
}

extern "C" void kernel_launch(void* const* d_in, const int* in_sizes, int n_in,
                              void* d_out, int out_size, void* d_ws, size_t ws_size,
                              hipStream_t stream) {
    (void)in_sizes; (void)n_in; (void)d_ws; (void)ws_size; (void)out_size;
    const float* parent = (const float*)d_in[0];  // parent_x
    // d_in[1] = leaf_x : unused (attention reduces to identity on V)
    const float* ck     = (const float*)d_in[2];  // conv_kernel (16,16)
    const float* bias   = (const float*)d_in[3];  // conv_bias (16,)
    // d_in[4] = Wq, d_in[5] = Wk : cancel out (softmax rows sum to 1)
    const float* wv     = (const float*)d_in[6];  // Wv (1,1)
    float* out          = (float*)d_out;

    const int threads = 256;                 // 8 waves per block
    const int blocks  = WAVES / 8;           // 4096 waves total
    cacomm_fused_kernel<<<blocks, threads, 0, stream>>>(parent, ck, bias, wv, out);
}